// PhraseClassifier_82386062672566
// MI455X (gfx1250) — compile-verified
//
#include <hip/hip_runtime.h>
#include <hip/hip_bf16.h>

typedef __attribute__((ext_vector_type(2))) float v2f;
typedef __attribute__((ext_vector_type(8))) float v8f;

#define SEQ     512
#define BATCH   64
#define DIM     512
#define NSPANS  100000
#define NGROUPS (NSPANS / 16)   // 6250, exact
#define FEAT    2048
#define EPSF    1e-12f

// ---------------- Kernel 1: pos_weight = N / sum(targets) - 1 ----------------
__global__ void pc_posweight_kernel(const float* __restrict__ targets,
                                    float* __restrict__ ws) {
    __shared__ float red[1024];
    int t = threadIdx.x;
    float s = 0.0f;
    for (int i = t; i < NSPANS; i += 1024) s += targets[i];   // fixed order per lane
    red[t] = s;
    __syncthreads();
    for (int off = 512; off > 0; off >>= 1) {                 // deterministic tree
        if (t < off) red[t] += red[t + off];
        __syncthreads();
    }
    if (t == 0) ws[0] = (float)NSPANS / red[0] - 1.0f;
}

// ---------------- Kernel 2: per-span logits via f32 WMMA + weighted BCE ------
// One wave handles 16 spans per group. A-operand = broadcast W chunk (every row
// of A equals W[4c..4c+3]); B-operand column n = span n's feature chunk. Every
// row of D then holds the 16 span logits, so each lane reads its span's logit
// from D element 0 (span index = lane % 16).
__global__ void pc_span_kernel(const float* __restrict__ hidden,
                               const float* __restrict__ W,
                               const float* __restrict__ bptr,
                               const float* __restrict__ targets,
                               const int*   __restrict__ span_bid,
                               const int*   __restrict__ span_begin,
                               const int*   __restrict__ span_end,
                               float* __restrict__ ws,
                               int totalWaves) {
    __shared__ __align__(16) float sW[FEAT];                  // 8 KB of W in LDS
    for (int i = threadIdx.x; i < FEAT; i += blockDim.x) sW[i] = W[i];
    __syncthreads();

    const int lane  = threadIdx.x & 31;
    const int wave  = blockIdx.x * (blockDim.x >> 5) + (threadIdx.x >> 5);
    const int half  = lane >> 4;        // 0: K={0,1}, 1: K={2,3}
    const int m     = lane & 15;        // span-in-group / D column index
    const float bias = bptr[0];
    const float posw = ws[0];

    float acc = 0.0f;                   // per-wave loss accumulator (lane 0 valid)

    for (int g = wave; g < NGROUPS; g += totalWaves) {
        const int s  = g * 16 + m;
        const int bb = span_bid[s];
        const int sb = span_begin[s];
        const int se = span_end[s];
        int pos[4] = { sb - 1, sb, se - 1, se };

        v8f c0 = {}; v8f c1 = {};
#pragma unroll
        for (int j = 0; j < 4; ++j) {
            const float* row  = hidden + (size_t)pos[j] * (BATCH * DIM)
                                       + (size_t)bb * DIM + 2 * half;
            const float* wseg = sW + (j * DIM + 2 * half);
#pragma unroll 4
            for (int c = 0; c < 128; c += 2) {
                v2f a0 = *(const v2f*)(wseg + 4 * c);
                v2f b0 = *(const v2f*)(row  + 4 * c);
                c0 = __builtin_amdgcn_wmma_f32_16x16x4_f32(
                        false, a0, false, b0, (short)0, c0, false, false);
                v2f a1 = *(const v2f*)(wseg + 4 * c + 4);
                v2f b1 = *(const v2f*)(row  + 4 * c + 4);
                c1 = __builtin_amdgcn_wmma_f32_16x16x4_f32(
                        false, a1, false, b1, (short)0, c1, false, false);
            }
        }

        const float logit = c0[0] + c1[0] + bias;
        float p = 1.0f / (1.0f + expf(-logit));
        p = fminf(fmaxf(p, EPSF), 1.0f - EPSF);
        const float tg   = targets[s];
        const float loss = -(tg * logf(p) + (1.0f - tg) * logf(1.0f - p));
        const float wgt  = (tg > 0.5f) ? posw : 1.0f;

        // lanes 0..15 and 16..31 hold duplicate spans: count lower half only
        float contrib = (half == 0) ? loss * wgt : 0.0f;
        for (int off = 16; off > 0; off >>= 1)
            contrib += __shfl_down(contrib, off, 32);
        acc += contrib;                  // valid on lane 0; no divergence in loop
    }

    if (lane == 0) ws[1 + wave] = acc;   // every wave writes its slot (0 if idle)
}

// ---------------- Kernel 3: reduce per-wave partials, divide by N ------------
__global__ void pc_finalize_kernel(const float* __restrict__ ws,
                                   float* __restrict__ out,
                                   int nPartials) {
    __shared__ float red[1024];
    int t = threadIdx.x;
    float s = 0.0f;
    for (int i = t; i < nPartials; i += 1024) s += ws[1 + i];
    red[t] = s;
    __syncthreads();
    for (int off = 512; off > 0; off >>= 1) {
        if (t < off) red[t] += red[t + off];
        __syncthreads();
    }
    if (t == 0) out[0] = red[0] / (float)NSPANS;
}

extern "C" void kernel_launch(void* const* d_in, const int* in_sizes, int n_in,
                              void* d_out, int out_size, void* d_ws, size_t ws_size,
                              hipStream_t stream) {
    const float* hidden  = (const float*)d_in[0];
    const float* W       = (const float*)d_in[1];
    const float* b       = (const float*)d_in[2];
    const float* targets = (const float*)d_in[3];
    const int* span_bid   = (const int*)d_in[4];
    const int* span_begin = (const int*)d_in[5];
    const int* span_end   = (const int*)d_in[6];
    float* out = (float*)d_out;
    float* ws  = (float*)d_ws;

    // ws layout: [0] = pos_weight, [1 .. totalWaves] = per-wave loss partials
    int totalWaves = 4096;                                   // 512 blocks x 8 waves
    long maxPartials = (long)(ws_size / sizeof(float)) - 1;
    if (maxPartials < totalWaves) {
        totalWaves = (int)((maxPartials / 8) * 8);
        if (totalWaves < 8) totalWaves = 8;
    }
    int blocks = totalWaves / 8;                             // 256 thr = 8 waves/block

    pc_posweight_kernel<<<1, 1024, 0, stream>>>(targets, ws);
    pc_span_kernel<<<blocks, 256, 0, stream>>>(hidden, W, b, targets,
                                               span_bid, span_begin, span_end,
                                               ws, totalWaves);
    pc_finalize_kernel<<<1, 1024, 0, stream>>>(ws, out, totalWaves);
}